// ACOPF_49306224558513
// MI455X (gfx1250) — compile-verified
//
#include <hip/hip_runtime.h>

// ---------------- problem constants ----------------
#define N_BUS   1000
#define N_LINE  2000
#define N_GEN   100
#define N_LOAD  500
#define NVM     900     // # Vm columns  (nongen count)
#define NPR     999     // # P rows      (nonref count)
#define NV      1899    // total unknowns / residuals
#define NP      1920    // padded: multiple of 128 (GEMM block) and 16 (WMMA)
#define N_COO   9000    // 4*N_LINE + N_BUS
#define RIDGE_F 1e-6f
#define CG_ITERS 400
#define GN_STEPS 10
#define BT      128     // GEMM block tile (output 128x128 per workgroup)
#define KT      16      // GEMM k-chunk staged in LDS

typedef float v2f __attribute__((ext_vector_type(2)));
typedef float v8f __attribute__((ext_vector_type(8)));

// ---------------- setup kernels ----------------

__global__ void k_init_maps(int* posGen, int* posLoad, int* posNongen, int* posNonref) {
  int b = blockIdx.x * blockDim.x + threadIdx.x;
  if (b < N_BUS) { posGen[b] = -1; posLoad[b] = -1; posNongen[b] = -1; posNonref[b] = -1; }
}

__global__ void k_fill_maps(const int* gen_buses, const int* load_buses,
                            const int* nongen_idx, const int* nonref_idx,
                            int* posGen, int* posLoad, int* posNongen, int* posNonref) {
  int t = blockIdx.x * blockDim.x + threadIdx.x;
  if (t < N_GEN)                       posGen[gen_buses[t]] = t;
  else if (t < N_GEN + N_LOAD)         posLoad[load_buses[t - N_GEN]] = t - N_GEN;
  else if (t < N_GEN + N_LOAD + NVM)   posNongen[nongen_idx[t - N_GEN - N_LOAD]] = t - N_GEN - N_LOAD;
  else if (t < N_GEN + N_LOAD + NVM + NPR)
                                       posNonref[nonref_idx[t - N_GEN - N_LOAD - NVM]] = t - N_GEN - N_LOAD - NVM;
}

__global__ void k_build_coo(const float* line_g, const float* line_b, const float* shunt_g,
                            const float* shunt_b, const float* tap, const float* shift,
                            const float* charging, const int* line_from, const int* line_to,
                            int* cooRow, int* cooCol, float* cooRe, float* cooIm) {
  int e = blockIdx.x * blockDim.x + threadIdx.x;
  if (e >= N_COO) return;
  int row, col; float re, im;
  if (e < 4 * N_LINE) {
    int l = e % N_LINE, w = e / N_LINE;
    float g = line_g[l], b = line_b[l], c = charging[l], tp = tap[l], sh = shift[l];
    float ysre = g, ysim = b;           // ys = g + jb
    float bcim = 0.5f * c;              // bc = j*c/2
    int f = line_from[l], t = line_to[l];
    float cs = cosf(sh), sn = sinf(sh);
    float invt = 1.0f / tp;
    if (w == 0) {                       // Yff = (ys+bc)/tap^2 at (f,f)
      re = ysre * invt * invt; im = (ysim + bcim) * invt * invt; row = f; col = f;
    } else if (w == 1) {                // Ytt = ys+bc at (t,t)
      re = ysre; im = ysim + bcim; row = t; col = t;
    } else if (w == 2) {                // Yft = -ys*e^{+j sh}/tap at (f,t)
      re = -(ysre * cs - ysim * sn) * invt;
      im = -(ysre * sn + ysim * cs) * invt;
      row = f; col = t;
    } else {                            // Ytf = -ys*e^{-j sh}/tap at (t,f)
      re = -(ysre * cs + ysim * sn) * invt;
      im =  (ysre * sn - ysim * cs) * invt;
      row = t; col = f;
    }
  } else {                              // shunt diagonal
    int b = e - 4 * N_LINE;
    row = b; col = b; re = shunt_g[b]; im = shunt_b[b];
  }
  cooRow[e] = row; cooCol[e] = col; cooRe[e] = re; cooIm[e] = im;
}

__global__ void k_init_x(float* x) {
  int i = blockIdx.x * blockDim.x + threadIdx.x;
  if (i < NP) x[i] = (i < NVM) ? 1.0f : 0.0f;
}

// ---------------- per-step kernels ----------------

__global__ void k_scatter_v(const float* x, const float* V_gen,
                            const int* posGen, const int* posNongen, const int* posNonref,
                            float* Vm, float* Va, float* Vre, float* Vim,
                            float* Ire, float* Iim) {
  int b = blockIdx.x * blockDim.x + threadIdx.x;
  if (b >= N_BUS) return;
  int qn = posNongen[b];
  float vm = (qn >= 0) ? x[qn] : V_gen[posGen[b]];
  int pr = posNonref[b];
  float va = (pr >= 0) ? x[NVM + pr] : 0.0f;
  Vm[b] = vm; Va[b] = va;
  Vre[b] = vm * cosf(va);
  Vim[b] = vm * sinf(va);
  Ire[b] = 0.0f; Iim[b] = 0.0f;
}

__global__ void k_accum_current(const int* cooRow, const int* cooCol,
                                const float* cooRe, const float* cooIm,
                                const float* Vre, const float* Vim,
                                float* Ire, float* Iim) {
  int e = blockIdx.x * blockDim.x + threadIdx.x;
  if (e >= N_COO) return;
  int i = cooRow[e], k = cooCol[e];
  float yre = cooRe[e], yim = cooIm[e];
  float vre = Vre[k],  vim = Vim[k];
  atomicAdd(&Ire[i], yre * vre - yim * vim);
  atomicAdd(&Iim[i], yre * vim + yim * vre);
}

__global__ void k_residual(const float* Vre, const float* Vim, const float* Ire, const float* Iim,
                           const float* P_gen, const float* P_load, const float* Q_load,
                           const int* posGen, const int* posLoad,
                           const int* posNonref, const int* posNongen,
                           float* Sre, float* Sim, float* Padj, float* Qadj, float* r) {
  int b = blockIdx.x * blockDim.x + threadIdx.x;
  if (b >= N_BUS) return;
  float vre = Vre[b], vim = Vim[b], ire = Ire[b], iim = Iim[b];
  float sre = vre * ire + vim * iim;     // Re(V conj(I))
  float sim = vim * ire - vre * iim;     // Im(V conj(I))
  Sre[b] = sre; Sim[b] = sim;
  float p = sre, q = sim;
  int pg = posGen[b];  if (pg >= 0) p -= P_gen[pg];
  int pl = posLoad[b]; if (pl >= 0) { p -= P_load[pl]; q -= Q_load[pl]; }
  Padj[b] = p; Qadj[b] = q;
  int pr = posNonref[b]; if (pr >= 0) r[pr] = p;
  int qn = posNongen[b]; if (qn >= 0) r[NPR + qn] = q;
}

__global__ void k_zero(float* p, int n) {
  int stride = gridDim.x * blockDim.x;
  for (int i = blockIdx.x * blockDim.x + threadIdx.x; i < n; i += stride) p[i] = 0.0f;
}

__global__ void k_build_J(const int* cooRow, const int* cooCol,
                          const float* cooRe, const float* cooIm,
                          const float* Vre, const float* Vim, const float* Vm,
                          const float* Sre, const float* Sim,
                          const int* posNonref, const int* posNongen, float* J) {
  int e = blockIdx.x * blockDim.x + threadIdx.x;
  if (e >= N_COO + N_BUS) return;
  int i, k;
  float dPdA, dQdA, dPdM, dQdM;
  if (e < N_COO) {
    i = cooRow[e]; k = cooCol[e];
    float yre = cooRe[e], yim = cooIm[e];
    float vkre = Vre[k], vkim = Vim[k];
    float tre = yre * vkre - yim * vkim;     // t = Y*V_k
    float tim = yre * vkim + yim * vkre;
    float vire = Vre[i], viim = Vim[i];
    float wre = vire * tre + viim * tim;     // w = V_i * conj(t)
    float wim = viim * tre - vire * tim;
    float vmk = Vm[k];
    float inv = (vmk != 0.0f) ? 1.0f / vmk : 0.0f;
    dPdA = wim;       dQdA = -wre;           // dS_i/dtheta_k = -j*w
    dPdM = wre * inv; dQdM = wim * inv;      // dS_i/dVm_k   =  w/Vm_k
  } else {
    int b = e - N_COO;                       // diagonal extra: j*V*conj(I), V*conj(I)/Vm
    i = b; k = b;
    float sre = Sre[b], sim = Sim[b];
    float vmb = Vm[b];
    float inv = (vmb != 0.0f) ? 1.0f / vmb : 0.0f;
    dPdA = -sim;      dQdA = sre;
    dPdM = sre * inv; dQdM = sim * inv;
  }
  int pr = posNonref[i];          // P row
  int qr = posNongen[i];          // Q row (offset NPR)
  int ca = posNonref[k];          // Va col (offset NVM)
  int cm = posNongen[k];          // Vm col
  if (pr >= 0) {
    float* Jr = J + (size_t)pr * NP;
    if (cm >= 0) atomicAdd(Jr + cm, dPdM);
    if (ca >= 0) atomicAdd(Jr + NVM + ca, dPdA);
  }
  if (qr >= 0) {
    float* Jr = J + (size_t)(NPR + qr) * NP;
    if (cm >= 0) atomicAdd(Jr + cm, dQdM);
    if (ca >= 0) atomicAdd(Jr + NVM + ca, dQdA);
  }
}

// g = J^T r  (padded columns -> 0)
__global__ void k_jtr(const float* __restrict__ J, const float* __restrict__ r, float* g) {
  int c = blockIdx.x * blockDim.x + threadIdx.x;
  if (c >= NP) return;
  if (c >= NV) { g[c] = 0.0f; return; }
  float s = 0.0f;
  for (int row = 0; row < NV; ++row) s += J[(size_t)row * NP + c] * r[row];
  g[c] = s;
}

// ---------------------------------------------------------------------------
// M = J^T J + ridge*I  --  LDS-tiled WMMA GEMM.
// Workgroup (8 waves) computes a 128x128 tile; wave w owns rows [16w,16w+16).
// Panels A = J[r0:r0+16, a0:a0+128], B = J[r0:r0+16, b0:b0+128] are staged in
// LDS via GLOBAL_LOAD_ASYNC_TO_LDS_B128 (ASYNCcnt, GVS addressing), then each
// wave reuses one 16x4 f32 A-fragment across 8 B-fragments / 8 WMMAs.
// f32 A 16x4 layout: lanes 0-15 hold K=0,1; lanes 16-31 hold K=2,3.
// f32 B 4x16 layout: v[0] = K rows 0/2, v[1] = K rows 1/3 (lane halves).
// f32 C 16x16 layout: VGPR v -> M=v (lanes 0-15), M=v+8 (lanes 16-31); N=lane&15.
// ---------------------------------------------------------------------------
__global__ __launch_bounds__(256) void k_jtj_tiled(const float* __restrict__ J,
                                                   float* __restrict__ M) {
  __shared__ float As[KT][BT];
  __shared__ float Bs[KT][BT];
  const int tilesPerDim = NP / BT;                 // 15
  int ta = blockIdx.x / tilesPerDim;
  int tb = blockIdx.x % tilesPerDim;
  int a0 = ta * BT, b0 = tb * BT;
  int tid  = threadIdx.x;
  int wave = tid >> 5;
  int lane = tid & 31;
  int m = lane & 15;
  int khalf = lane >> 4;                           // 0 -> K base 0, 1 -> K base 2
  unsigned long long jbase = (unsigned long long)J;
  unsigned asBase = (unsigned)(uintptr_t)&As[0][0];
  unsigned bsBase = (unsigned)(uintptr_t)&Bs[0][0];

  v8f zero = {};
  v8f acc[8];
#pragma unroll
  for (int s2 = 0; s2 < 8; ++s2) acc[s2] = zero;

  for (int r0 = 0; r0 < NP; r0 += KT) {
    // stage both 16x128 panels into LDS with async DMA loads (2 b128 slots/thread/panel)
#pragma unroll
    for (int s = 0; s < 2; ++s) {
      int slot = tid + s * 256;                    // 512 b128 slots per panel
      int row  = slot >> 5;                        // 32 float4 per 128-wide row
      int c4   = slot & 31;
      unsigned goffA = (unsigned)((((r0 + row) * NP) + a0 + c4 * 4) * 4);
      unsigned goffB = (unsigned)((((r0 + row) * NP) + b0 + c4 * 4) * 4);
      unsigned la = asBase + (unsigned)(slot * 16);
      unsigned lb = bsBase + (unsigned)(slot * 16);
      asm volatile("global_load_async_to_lds_b128 %0, %1, %2"
                   :: "v"(la), "v"(goffA), "s"(jbase) : "memory");
      asm volatile("global_load_async_to_lds_b128 %0, %1, %2"
                   :: "v"(lb), "v"(goffB), "s"(jbase) : "memory");
    }
    asm volatile("s_wait_asynccnt 0x0" ::: "memory");
    __syncthreads();

#pragma unroll
    for (int kk = 0; kk < KT; kk += 4) {
      v2f a;
      a.x = As[kk + khalf * 2][wave * 16 + m];
      a.y = As[kk + khalf * 2 + 1][wave * 16 + m];
#pragma unroll
      for (int sub = 0; sub < 8; ++sub) {
        v2f b;
        b.x = Bs[kk + khalf * 2][sub * 16 + m];
        b.y = Bs[kk + khalf * 2 + 1][sub * 16 + m];
        acc[sub] = __builtin_amdgcn_wmma_f32_16x16x4_f32(
            /*neg_a=*/false, a, /*neg_b=*/false, b,
            /*c_mod=*/(short)0, acc[sub], /*reuse_a=*/false, /*reuse_b=*/false);
      }
    }
    __syncthreads();
  }

#pragma unroll
  for (int sub = 0; sub < 8; ++sub) {
#pragma unroll
    for (int v = 0; v < 8; ++v) {
      int row = a0 + wave * 16 + v + khalf * 8;
      int col = b0 + sub * 16 + m;
      float val = acc[sub][v];
      if (row == col) val += (row < NV) ? RIDGE_F : 1.0f;   // ridge; unit diag in padding
      M[(size_t)row * NP + col] = val;
    }
  }
}

// single-workgroup fixed-iteration CG: solve M*dx = g (SPD, NPxNP)
__device__ __forceinline__ float block_reduce_1024(float v, float* red) {
  int tid = threadIdx.x;
  red[tid] = v; __syncthreads();
  for (int s = 512; s > 0; s >>= 1) {
    if (tid < s) red[tid] += red[tid + s];
    __syncthreads();
  }
  float r = red[0]; __syncthreads();
  return r;
}

__global__ __launch_bounds__(1024) void k_cg_solve(const float* __restrict__ M,
                                                   const float* __restrict__ g,
                                                   float* dx, float* rv, float* pv, float* qv) {
  __shared__ float red[1024];
  __shared__ float sp[NP];
  int tid = threadIdx.x;
  for (int i = tid; i < NP; i += 1024) { dx[i] = 0.0f; float gi = g[i]; rv[i] = gi; pv[i] = gi; }
  __syncthreads();
  float part = 0.0f;
  for (int i = tid; i < NP; i += 1024) part += rv[i] * rv[i];
  float rr = block_reduce_1024(part, red);
  for (int it = 0; it < CG_ITERS; ++it) {
    for (int i = tid; i < NP; i += 1024) sp[i] = pv[i];
    __syncthreads();
    for (int row = tid; row < NP; row += 1024) {
      const float* Mr = M + (size_t)row * NP;
      float s = 0.0f;
      for (int c2 = 0; c2 < NP; ++c2) s += Mr[c2] * sp[c2];
      qv[row] = s;
    }
    __syncthreads();
    part = 0.0f;
    for (int i = tid; i < NP; i += 1024) part += sp[i] * qv[i];
    float pAp = block_reduce_1024(part, red);
    float alpha = (pAp > 1e-30f) ? rr / pAp : 0.0f;
    for (int i = tid; i < NP; i += 1024) { dx[i] += alpha * sp[i]; rv[i] -= alpha * qv[i]; }
    __syncthreads();
    part = 0.0f;
    for (int i = tid; i < NP; i += 1024) part += rv[i] * rv[i];
    float rrn = block_reduce_1024(part, red);
    float beta = (rr > 1e-30f) ? rrn / rr : 0.0f;
    for (int i = tid; i < NP; i += 1024) pv[i] = rv[i] + beta * sp[i];
    __syncthreads();
    rr = rrn;
  }
}

__global__ void k_update_x(float* x, const float* dx) {
  int i = blockIdx.x * blockDim.x + threadIdx.x;
  if (i < NV) x[i] -= dx[i];
}

// ---------------- output kernels ----------------

__global__ void k_res_norm(const float* r, float* out_scalar) {
  __shared__ float red[256];
  float s = 0.0f;
  for (int i = threadIdx.x; i < NV; i += 256) s += r[i] * r[i];
  red[threadIdx.x] = s; __syncthreads();
  for (int st = 128; st > 0; st >>= 1) {
    if (threadIdx.x < st) red[threadIdx.x] += red[threadIdx.x + st];
    __syncthreads();
  }
  if (threadIdx.x == 0) out_scalar[0] = red[0];
}

__global__ void k_outputs(const float* Vm, const float* Va,
                          const float* Padj, const float* Qadj,
                          const int* posGen, const int* refp,
                          const float* bus_v_limits, float* out) {
  int b = blockIdx.x * blockDim.x + threadIdx.x;
  if (b >= N_BUS) return;
  float vm = Vm[b];
  out[b] = vm;                         // Vm [0,1000)
  out[N_BUS + b] = Va[b];              // Va [1000,2000)
  int pg = posGen[b];
  if (pg >= 0) out[2 * N_BUS + pg] = Qadj[b];          // Q_gen [2000,2100)
  if (b == refp[0]) {
    out[2 * N_BUS + N_GEN]     = Padj[b];              // P_ref at 2100
    out[2 * N_BUS + N_GEN + 1] = Qadj[b];              // Q_ref at 2101
  }
  float vmin = bus_v_limits[2 * b], vmax = bus_v_limits[2 * b + 1];
  out[2 * N_BUS + N_GEN + 3 + b] =                     // V_violation [2103,3103)
      fmaxf(vm - vmax, 0.0f) + fmaxf(vmin - vm, 0.0f);
}

// ---------------- host driver ----------------

extern "C" void kernel_launch(void* const* d_in, const int* in_sizes, int n_in,
                              void* d_out, int out_size, void* d_ws, size_t ws_size,
                              hipStream_t stream) {
  (void)in_sizes; (void)n_in; (void)out_size; (void)ws_size;
  const float* line_g    = (const float*)d_in[0];
  const float* line_b    = (const float*)d_in[1];
  const float* shunt_g   = (const float*)d_in[2];
  const float* shunt_b   = (const float*)d_in[3];
  const float* tap       = (const float*)d_in[4];
  const float* shift     = (const float*)d_in[5];
  const float* charging  = (const float*)d_in[6];
  const float* P_gen     = (const float*)d_in[7];
  const float* V_gen     = (const float*)d_in[8];
  const float* P_load    = (const float*)d_in[9];
  const float* Q_load    = (const float*)d_in[10];
  const float* vlimits   = (const float*)d_in[11];
  const int*   line_from = (const int*)d_in[12];
  const int*   line_to   = (const int*)d_in[13];
  const int*   gen_buses = (const int*)d_in[14];
  const int*   load_buses= (const int*)d_in[15];
  const int*   nongen_idx= (const int*)d_in[16];
  const int*   nonref_idx= (const int*)d_in[17];
  const int*   refp      = (const int*)d_in[18];
  float* out = (float*)d_out;

  // workspace carve-up (256B aligned)
  char* base = (char*)d_ws;
  size_t off = 0;
  auto alloc = [&](size_t bytes) -> void* {
    void* p = base + off;
    off += (bytes + 255) & ~(size_t)255;
    return p;
  };
  float* J    = (float*)alloc((size_t)NP * NP * sizeof(float));
  float* M    = (float*)alloc((size_t)NP * NP * sizeof(float));
  float* cooRe = (float*)alloc(N_COO * sizeof(float));
  float* cooIm = (float*)alloc(N_COO * sizeof(float));
  int*   cooRow = (int*)alloc(N_COO * sizeof(int));
  int*   cooCol = (int*)alloc(N_COO * sizeof(int));
  int*   posGen    = (int*)alloc(N_BUS * sizeof(int));
  int*   posLoad   = (int*)alloc(N_BUS * sizeof(int));
  int*   posNongen = (int*)alloc(N_BUS * sizeof(int));
  int*   posNonref = (int*)alloc(N_BUS * sizeof(int));
  float* Vm  = (float*)alloc(N_BUS * sizeof(float));
  float* Va  = (float*)alloc(N_BUS * sizeof(float));
  float* Vre = (float*)alloc(N_BUS * sizeof(float));
  float* Vim = (float*)alloc(N_BUS * sizeof(float));
  float* Ire = (float*)alloc(N_BUS * sizeof(float));
  float* Iim = (float*)alloc(N_BUS * sizeof(float));
  float* Sre = (float*)alloc(N_BUS * sizeof(float));
  float* Sim = (float*)alloc(N_BUS * sizeof(float));
  float* Padj = (float*)alloc(N_BUS * sizeof(float));
  float* Qadj = (float*)alloc(N_BUS * sizeof(float));
  float* x   = (float*)alloc(NP * sizeof(float));
  float* r   = (float*)alloc(NP * sizeof(float));
  float* g   = (float*)alloc(NP * sizeof(float));
  float* dx  = (float*)alloc(NP * sizeof(float));
  float* cgR = (float*)alloc(NP * sizeof(float));
  float* cgP = (float*)alloc(NP * sizeof(float));
  float* cgQ = (float*)alloc(NP * sizeof(float));

  const int B = 256;
  const int gBus  = (N_BUS + B - 1) / B;
  const int gCoo  = (N_COO + B - 1) / B;
  const int gNP   = (NP + B - 1) / B;
  const int gMaps = (N_GEN + N_LOAD + NVM + NPR + B - 1) / B;
  const int gBldJ = (N_COO + N_BUS + B - 1) / B;
  const int gGemm = (NP / BT) * (NP / BT);               // 15*15 = 225 workgroups

  // one-time setup (deterministic; rebuilt every call)
  k_init_maps<<<gBus, B, 0, stream>>>(posGen, posLoad, posNongen, posNonref);
  k_fill_maps<<<gMaps, B, 0, stream>>>(gen_buses, load_buses, nongen_idx, nonref_idx,
                                       posGen, posLoad, posNongen, posNonref);
  k_build_coo<<<gCoo, B, 0, stream>>>(line_g, line_b, shunt_g, shunt_b, tap, shift, charging,
                                      line_from, line_to, cooRow, cooCol, cooRe, cooIm);
  k_init_x<<<gNP, B, 0, stream>>>(x);

  for (int step = 0; step < GN_STEPS; ++step) {
    k_scatter_v<<<gBus, B, 0, stream>>>(x, V_gen, posGen, posNongen, posNonref,
                                        Vm, Va, Vre, Vim, Ire, Iim);
    k_accum_current<<<gCoo, B, 0, stream>>>(cooRow, cooCol, cooRe, cooIm, Vre, Vim, Ire, Iim);
    k_residual<<<gBus, B, 0, stream>>>(Vre, Vim, Ire, Iim, P_gen, P_load, Q_load,
                                       posGen, posLoad, posNonref, posNongen,
                                       Sre, Sim, Padj, Qadj, r);
    k_zero<<<1024, B, 0, stream>>>(J, NP * NP);
    k_build_J<<<gBldJ, B, 0, stream>>>(cooRow, cooCol, cooRe, cooIm, Vre, Vim, Vm,
                                       Sre, Sim, posNonref, posNongen, J);
    k_jtr<<<gNP, B, 0, stream>>>(J, r, g);
    k_jtj_tiled<<<gGemm, B, 0, stream>>>(J, M);
    k_cg_solve<<<1, 1024, 0, stream>>>(M, g, dx, cgR, cgP, cgQ);
    k_update_x<<<gNP, B, 0, stream>>>(x, dx);
  }

  // final evaluation at converged x
  k_scatter_v<<<gBus, B, 0, stream>>>(x, V_gen, posGen, posNongen, posNonref,
                                      Vm, Va, Vre, Vim, Ire, Iim);
  k_accum_current<<<gCoo, B, 0, stream>>>(cooRow, cooCol, cooRe, cooIm, Vre, Vim, Ire, Iim);
  k_residual<<<gBus, B, 0, stream>>>(Vre, Vim, Ire, Iim, P_gen, P_load, Q_load,
                                     posGen, posLoad, posNonref, posNongen,
                                     Sre, Sim, Padj, Qadj, r);
  k_res_norm<<<1, 256, 0, stream>>>(r, out + 2 * N_BUS + N_GEN + 2);   // index 2102
  k_outputs<<<gBus, B, 0, stream>>>(Vm, Va, Padj, Qadj, posGen, refp, vlimits, out);
}